// CCCA_42331197669734
// MI455X (gfx1250) — compile-verified
//
#include <hip/hip_runtime.h>

// ---------------------------------------------------------------------------
// CCCA (criss-cross attention + SE gate) for MI455X / gfx1250, fp32 WMMA path.
// Round 2: fused SE pooling into the column-attention kernel (removes one full
// 151 MB read of O) + global_prefetch hints on streamed operands.
// ---------------------------------------------------------------------------

typedef float v2f __attribute__((ext_vector_type(2)));
typedef float v8f __attribute__((ext_vector_type(8)));

#define WMMA4(a, b, c) \
  __builtin_amdgcn_wmma_f32_16x16x4_f32(false, (a), false, (b), (short)0, (c), false, false)

constexpr int NB  = 16;        // batch
constexpr int CH  = 256;       // channels
constexpr int HH  = 96;
constexpr int WW  = 96;
constexpr int PP  = HH * WW;   // 9216 spatial positions
constexpr int QKD = 4;         // qk projection dim
constexpr int SSTR = 98;       // padded LDS row stride for 96-wide score rows

// Order-preserving float<->uint key (for atomicMax on floats via uint).
__device__ __forceinline__ unsigned f2key(float f) {
  unsigned u = __float_as_uint(f);
  return (u & 0x80000000u) ? ~u : (u | 0x80000000u);
}
__device__ __forceinline__ float key2f(unsigned k) {
  return (k & 0x80000000u) ? __uint_as_float(k ^ 0x80000000u) : __uint_as_float(~k);
}

// ---------------------------------------------------------------------------
// Kernel 0: init fused-pool accumulators (ws is poisoned; must zero each call).
// ---------------------------------------------------------------------------
__global__ void k_init(float* __restrict__ psum, unsigned* __restrict__ pmax)
{
  int i = blockIdx.x * blockDim.x + threadIdx.x;
  if (i < NB * CH) {
    psum[i] = 0.f;
    pmax[i] = 0u;      // below f2key(-inf) = 0x007FFFFF
  }
}

// ---------------------------------------------------------------------------
// Kernel 1: fused V = Wv @ x  (WMMA)  and  Q/K = Wq/Wk @ x (VALU piggyback).
// Grid: (PP/64, NB), block 256 (8 waves). Wave w computes V rows [w*32, w*32+32)
// for a 64-wide spatial tile.
// ---------------------------------------------------------------------------
__global__ __launch_bounds__(256) void k_proj(
    const float* __restrict__ x,  const float* __restrict__ Wq,
    const float* __restrict__ Wk, const float* __restrict__ Wv,
    float* __restrict__ Vout, float* __restrict__ Qout, float* __restrict__ Kout)
{
  __shared__ float xs[16 * 64];    // x chunk  [c:16][p:64]
  __shared__ float wvs[256 * 16];  // Wv chunk [o:256][c:16]

  const int tid  = threadIdx.x;
  const int lane = tid & 31;
  const int wave = tid >> 5;
  const int n    = blockIdx.y;
  const int p0   = blockIdx.x * 64;
  const int krow = 2 * (lane >> 4);   // 0 or 2
  const int lc   = lane & 15;

  v8f acc[2][4];
#pragma unroll
  for (int m = 0; m < 2; ++m)
#pragma unroll
    for (int t = 0; t < 4; ++t)
#pragma unroll
      for (int r = 0; r < 8; ++r) acc[m][t][r] = 0.f;

  float qa[QKD] = {0.f, 0.f, 0.f, 0.f};
  float ka[QKD] = {0.f, 0.f, 0.f, 0.f};

  const float* xb = x + (size_t)(n * CH) * PP;

  for (int c0 = 0; c0 < CH; c0 += 16) {
    __syncthreads();
    // stage x[c0:c0+16][p0:p0+64]  (coalesced: contiguous p within a row)
#pragma unroll
    for (int i = 0; i < 4; ++i) {
      int idx = tid + i * 256;
      int cc = idx >> 6, pp_ = idx & 63;
      xs[idx] = xb[(size_t)(c0 + cc) * PP + p0 + pp_];
    }
    // stage Wv[0:256][c0:c0+16]  (16 contiguous floats per thread-row)
#pragma unroll
    for (int j = 0; j < 16; ++j) wvs[tid * 16 + j] = Wv[tid * CH + c0 + j];
    // prefetch next x chunk rows (global_prefetch_b8)
    if (c0 + 16 < CH && tid < 16)
      __builtin_prefetch(xb + (size_t)(c0 + 16 + tid) * PP + p0, 0, 1);
    __syncthreads();

#pragma unroll
    for (int kk = 0; kk < 4; ++kk) {
      v2f a[2], b[4];
#pragma unroll
      for (int m = 0; m < 2; ++m) {
        int row = wave * 32 + m * 16 + lc;
        a[m][0] = wvs[row * 16 + 4 * kk + krow + 0];
        a[m][1] = wvs[row * 16 + 4 * kk + krow + 1];
      }
#pragma unroll
      for (int t = 0; t < 4; ++t) {
        int col = t * 16 + lc;
        b[t][0] = xs[(4 * kk + krow + 0) * 64 + col];
        b[t][1] = xs[(4 * kk + krow + 1) * 64 + col];
      }
#pragma unroll
      for (int m = 0; m < 2; ++m)
#pragma unroll
        for (int t = 0; t < 4; ++t) acc[m][t] = WMMA4(a[m], b[t], acc[m][t]);
    }

    // Q/K projection piggyback: waves 0,1 cover the 64 spatial positions.
    if (wave < 2) {
      int pl = wave * 32 + lane;
#pragma unroll
      for (int cc = 0; cc < 16; ++cc) {
        float xv = xs[cc * 64 + pl];
        int c = c0 + cc;
#pragma unroll
        for (int q = 0; q < QKD; ++q) {
          qa[q] += Wq[q * CH + c] * xv;   // lane-uniform weight index -> s_load
          ka[q] += Wk[q * CH + c] * xv;
        }
      }
    }
  }

  float* vb = Vout + (size_t)(n * CH) * PP;
#pragma unroll
  for (int m = 0; m < 2; ++m)
#pragma unroll
    for (int t = 0; t < 4; ++t)
#pragma unroll
      for (int r = 0; r < 8; ++r) {
        int o = wave * 32 + m * 16 + r + 8 * (lane >> 4);
        vb[(size_t)o * PP + p0 + t * 16 + lc] = acc[m][t][r];
      }

  if (wave < 2) {
    int pl = wave * 32 + lane;
#pragma unroll
    for (int q = 0; q < QKD; ++q) {
      Qout[(size_t)(n * QKD + q) * PP + p0 + pl] = qa[q];
      Kout[(size_t)(n * QKD + q) * PP + p0 + pl] = ka[q];
    }
  }
}

// ---------------------------------------------------------------------------
// Kernel 2: row (width-axis) attention for one (n,h) slice.
//   S[w][v] = sum_q Q[q][w]*K[q][v]; softmax over v; out1[c][w] = sum_v V[c][v]*A[w][v]
// Writes O (= d_out) with out1. Grid: (HH, NB), block 256.
// ---------------------------------------------------------------------------
__global__ __launch_bounds__(256) void k_row_attn(
    const float* __restrict__ Qw, const float* __restrict__ Kw,
    const float* __restrict__ Vw, float* __restrict__ O)
{
  __shared__ float qs[QKD * WW];    // [q][w]
  __shared__ float ks[QKD * WW];    // [q][v]
  __shared__ float ss[WW * SSTR];   // scores -> attention [w][v]

  const int tid  = threadIdx.x;
  const int lane = tid & 31;
  const int wave = tid >> 5;
  const int n    = blockIdx.y;
  const int h    = blockIdx.x;
  const int krow = 2 * (lane >> 4);
  const int lc   = lane & 15;

  const float* vb = Vw + (size_t)(n * CH) * PP + (size_t)h * WW;   // V[n][c][h][v]

  for (int idx = tid; idx < QKD * WW; idx += 256) {
    int q = idx / WW, w = idx % WW;
    qs[idx] = Qw[(size_t)(n * QKD + q) * PP + h * WW + w];
    ks[idx] = Kw[(size_t)(n * QKD + q) * PP + h * WW + w];
  }
  // prefetch the V slice rows we will stream in the apply phase
  __builtin_prefetch(vb + (size_t)tid * PP, 0, 1);
  __syncthreads();

  // ---- scores: 36 tiles of 16x16, K=4 (one WMMA each), spread over 8 waves
  for (int tt = wave; tt < 36; tt += 8) {
    int mi = tt / 6, ni = tt % 6;
    v2f a, b;
    a[0] = qs[(krow + 0) * WW + mi * 16 + lc];
    a[1] = qs[(krow + 1) * WW + mi * 16 + lc];
    b[0] = ks[(krow + 0) * WW + ni * 16 + lc];
    b[1] = ks[(krow + 1) * WW + ni * 16 + lc];
    v8f cz;
#pragma unroll
    for (int r = 0; r < 8; ++r) cz[r] = 0.f;
    v8f d = WMMA4(a, b, cz);
#pragma unroll
    for (int r = 0; r < 8; ++r)
      ss[(mi * 16 + r + 8 * (lane >> 4)) * SSTR + ni * 16 + lc] = d[r];
  }
  __syncthreads();

  // ---- softmax over v, one thread per row w (96 = 3 full waves -> uniform)
  if (tid < WW) {
    float mx = -3.402823e38f;
    for (int v = 0; v < WW; ++v) mx = fmaxf(mx, ss[tid * SSTR + v]);
    float s = 0.f;
    for (int v = 0; v < WW; ++v) {
      float e = __expf(ss[tid * SSTR + v] - mx);
      ss[tid * SSTR + v] = e;
      s += e;
    }
    float inv = 1.f / s;
    for (int v = 0; v < WW; ++v) ss[tid * SSTR + v] *= inv;
  }
  __syncthreads();

  // ---- apply: out1[c][w], M=256 (wave handles 32 rows), N=96, K=96
  v8f acc[2][6];
#pragma unroll
  for (int m = 0; m < 2; ++m)
#pragma unroll
    for (int t = 0; t < 6; ++t)
#pragma unroll
      for (int r = 0; r < 8; ++r) acc[m][t][r] = 0.f;

  for (int v0 = 0; v0 < WW; v0 += 4) {
    v2f a[2], b[6];
#pragma unroll
    for (int m = 0; m < 2; ++m) {
      int c = wave * 32 + m * 16 + lc;
      const float* pv = vb + (size_t)c * PP + v0 + krow;
      a[m][0] = pv[0];
      a[m][1] = pv[1];
    }
#pragma unroll
    for (int t = 0; t < 6; ++t) {
      int w = t * 16 + lc;
      b[t][0] = ss[w * SSTR + v0 + krow + 0];
      b[t][1] = ss[w * SSTR + v0 + krow + 1];
    }
#pragma unroll
    for (int m = 0; m < 2; ++m)
#pragma unroll
      for (int t = 0; t < 6; ++t) acc[m][t] = WMMA4(a[m], b[t], acc[m][t]);
  }

  float* ob = O + (size_t)(n * CH) * PP + (size_t)h * WW;   // O[n][c][h][w]
#pragma unroll
  for (int m = 0; m < 2; ++m)
#pragma unroll
    for (int t = 0; t < 6; ++t)
#pragma unroll
      for (int r = 0; r < 8; ++r) {
        int c = wave * 32 + m * 16 + r + 8 * (lane >> 4);
        ob[(size_t)c * PP + t * 16 + lc] = acc[m][t][r];
      }
}

// ---------------------------------------------------------------------------
// Kernel 3: column (height-axis) attention for one (n,w) slice; O += out2,
// and FUSED SE pooling (sum & max over spatial) via atomics, since this kernel
// produces the final pre-gate value of O for its column.
// ---------------------------------------------------------------------------
__global__ __launch_bounds__(256) void k_col_attn(
    const float* __restrict__ Qw, const float* __restrict__ Kw,
    const float* __restrict__ Vw, float* __restrict__ O,
    float* __restrict__ psum, unsigned* __restrict__ pmax)
{
  __shared__ float qs[QKD * HH];    // [q][h]
  __shared__ float ks[QKD * HH];    // [q][g]
  __shared__ float ss[HH * SSTR];   // [h][g]

  const int tid  = threadIdx.x;
  const int lane = tid & 31;
  const int wave = tid >> 5;
  const int n    = blockIdx.y;
  const int wc   = blockIdx.x;      // fixed w column
  const int krow = 2 * (lane >> 4);
  const int lc   = lane & 15;

  const float* vb = Vw + (size_t)(n * CH) * PP + wc;    // V[n][c][g][wc]

  for (int idx = tid; idx < QKD * HH; idx += 256) {
    int q = idx / HH, g = idx % HH;
    qs[idx] = Qw[(size_t)(n * QKD + q) * PP + g * WW + wc];
    ks[idx] = Kw[(size_t)(n * QKD + q) * PP + g * WW + wc];
  }
  __builtin_prefetch(vb + (size_t)tid * PP, 0, 1);
  __syncthreads();

  for (int tt = wave; tt < 36; tt += 8) {
    int mi = tt / 6, ni = tt % 6;
    v2f a, b;
    a[0] = qs[(krow + 0) * HH + mi * 16 + lc];
    a[1] = qs[(krow + 1) * HH + mi * 16 + lc];
    b[0] = ks[(krow + 0) * HH + ni * 16 + lc];
    b[1] = ks[(krow + 1) * HH + ni * 16 + lc];
    v8f cz;
#pragma unroll
    for (int r = 0; r < 8; ++r) cz[r] = 0.f;
    v8f d = WMMA4(a, b, cz);
#pragma unroll
    for (int r = 0; r < 8; ++r)
      ss[(mi * 16 + r + 8 * (lane >> 4)) * SSTR + ni * 16 + lc] = d[r];
  }
  __syncthreads();

  if (tid < HH) {
    float mx = -3.402823e38f;
    for (int g = 0; g < HH; ++g) mx = fmaxf(mx, ss[tid * SSTR + g]);
    float s = 0.f;
    for (int g = 0; g < HH; ++g) {
      float e = __expf(ss[tid * SSTR + g] - mx);
      ss[tid * SSTR + g] = e;
      s += e;
    }
    float inv = 1.f / s;
    for (int g = 0; g < HH; ++g) ss[tid * SSTR + g] *= inv;
  }
  __syncthreads();

  v8f acc[2][6];
#pragma unroll
  for (int m = 0; m < 2; ++m)
#pragma unroll
    for (int t = 0; t < 6; ++t)
#pragma unroll
      for (int r = 0; r < 8; ++r) acc[m][t][r] = 0.f;

  for (int g0 = 0; g0 < HH; g0 += 4) {
    v2f a[2], b[6];
#pragma unroll
    for (int m = 0; m < 2; ++m) {
      int c = wave * 32 + m * 16 + lc;
      a[m][0] = vb[(size_t)c * PP + (size_t)(g0 + krow + 0) * WW];
      a[m][1] = vb[(size_t)c * PP + (size_t)(g0 + krow + 1) * WW];
    }
#pragma unroll
    for (int t = 0; t < 6; ++t) {
      int hq = t * 16 + lc;
      b[t][0] = ss[hq * SSTR + g0 + krow + 0];
      b[t][1] = ss[hq * SSTR + g0 + krow + 1];
    }
#pragma unroll
    for (int m = 0; m < 2; ++m)
#pragma unroll
      for (int t = 0; t < 6; ++t) acc[m][t] = WMMA4(a[m], b[t], acc[m][t]);
  }

  // O += out2, producing the final pre-gate value; pool sum/max over h
  // in-register (shfl_xor over the 16-lane N-group of the D layout), then one
  // atomic per (c, half-wave).
  float* ob = O + (size_t)(n * CH) * PP + wc;
#pragma unroll
  for (int m = 0; m < 2; ++m)
#pragma unroll
    for (int r = 0; r < 8; ++r) {
      int c = wave * 32 + m * 16 + r + 8 * (lane >> 4);
      float s = 0.f, mx = -3.402823e38f;
#pragma unroll
      for (int t = 0; t < 6; ++t) {
        int hq = t * 16 + lc;
        size_t idx = (size_t)c * PP + (size_t)hq * WW;
        float fin = ob[idx] + acc[m][t][r];
        ob[idx] = fin;
        s += fin;
        mx = fmaxf(mx, fin);
      }
#pragma unroll
      for (int msk = 1; msk < 16; msk <<= 1) {
        s  += __shfl_xor(s, msk, 32);
        mx  = fmaxf(mx, __shfl_xor(mx, msk, 32));
      }
      if (lc == 0) {
        atomicAdd(&psum[n * CH + c], s);
        atomicMax(&pmax[n * CH + c], f2key(mx));
      }
    }
}

// ---------------------------------------------------------------------------
// Kernel 4: SE gate — sigmoid(relu(avg@W1^T)@W2^T + relu(max@W1^T)@W2^T).
// One block per n; thread j computes hidden j then output channel j.
// ---------------------------------------------------------------------------
__global__ __launch_bounds__(256) void k_gate(const float* __restrict__ psum,
                                              const unsigned* __restrict__ pmax,
                                              const float* __restrict__ W1,
                                              const float* __restrict__ W2,
                                              float* __restrict__ gate)
{
  __shared__ float sa_[CH];
  __shared__ float sm_[CH];
  __shared__ float hid[CH];
  const int n = blockIdx.x;
  const int j = threadIdx.x;
  sa_[j] = psum[n * CH + j] * (1.f / PP);   // avg pool
  sm_[j] = key2f(pmax[n * CH + j]);         // max pool
  __syncthreads();
  float sa = 0.f, sm = 0.f;
  for (int i = 0; i < CH; ++i) {
    float w = W1[j * CH + i];
    sa += w * sa_[i];
    sm += w * sm_[i];
  }
  hid[j] = fmaxf(sa, 0.f) + fmaxf(sm, 0.f);   // relu(a)@W2 + relu(m)@W2 = (ra+rm)@W2
  __syncthreads();
  float s = 0.f;
  for (int i = 0; i < CH; ++i) s += W2[j * CH + i] * hid[i];
  gate[n * CH + j] = 1.f / (1.f + __expf(-s));
}

// ---------------------------------------------------------------------------
// Kernel 5: final — out = gama * gate[n,c] * O + x   (in place on d_out).
// ---------------------------------------------------------------------------
__global__ void k_final(const float* __restrict__ x, const float* __restrict__ gate,
                        const float* __restrict__ gama, float* __restrict__ O)
{
  const size_t tot = (size_t)NB * CH * PP;
  const float g0 = gama[0];
  for (size_t i = (size_t)blockIdx.x * blockDim.x + threadIdx.x; i < tot;
       i += (size_t)gridDim.x * blockDim.x) {
    size_t row = i / PP;                 // n*CH + c
    O[i] = g0 * gate[row] * O[i] + x[i];
  }
}

// ---------------------------------------------------------------------------
extern "C" void kernel_launch(void* const* d_in, const int* in_sizes, int n_in,
                              void* d_out, int out_size, void* d_ws, size_t ws_size,
                              hipStream_t stream)
{
  (void)in_sizes; (void)n_in; (void)out_size; (void)ws_size;
  const float* x    = (const float*)d_in[0];
  const float* Wq   = (const float*)d_in[1];
  const float* Wk   = (const float*)d_in[2];
  const float* Wv   = (const float*)d_in[3];
  const float* W1   = (const float*)d_in[4];
  const float* W2   = (const float*)d_in[5];
  const float* gama = (const float*)d_in[6];
  float* O  = (float*)d_out;
  float* ws = (float*)d_ws;

  float*    V    = ws;                                   // NB*CH*PP
  float*    Q    = V + (size_t)NB * CH * PP;             // NB*QKD*PP
  float*    K    = Q + (size_t)NB * QKD * PP;            // NB*QKD*PP
  float*    psum = K + (size_t)NB * QKD * PP;            // NB*CH
  unsigned* pmax = (unsigned*)(psum + NB * CH);          // NB*CH
  float*    gate = (float*)(pmax + NB * CH);             // NB*CH

  k_init    <<<(NB * CH + 255) / 256, 256, 0, stream>>>(psum, pmax);
  k_proj    <<<dim3(PP / 64, NB), 256, 0, stream>>>(x, Wq, Wk, Wv, V, Q, K);
  k_row_attn<<<dim3(HH, NB),      256, 0, stream>>>(Q, K, V, O);
  k_col_attn<<<dim3(WW, NB),      256, 0, stream>>>(Q, K, V, O, psum, pmax);
  k_gate    <<<NB,                256, 0, stream>>>(psum, pmax, W1, W2, gate);
  k_final   <<<4096,              256, 0, stream>>>(x, gate, gama, O);
}